// Qadder2d_87608742904029
// MI455X (gfx1250) — compile-verified
//
#include <hip/hip_runtime.h>
#include <hip/hip_bf16.h>
#include <stdint.h>

// Problem constants (from reference): x[16,32,128,128] f32, w[32,32,3,3] f32,
// out[16,32,128,128] f32 = -(SAD of int8-quantized patch vs weight) * 2^-7.
#define NB   16
#define CIN  32
#define HH   128
#define WW   128
#define COUT 32
#define QX_BYTES ((size_t)NB * HH * WW * CIN)      // 8,388,608 u8 (NHWC)
#define QW_BYTES ((size_t)COUT * 9 * CIN)          // 9,216 u8  [cout][kpos][cin]

static __device__ __forceinline__ unsigned sad8(unsigned a, unsigned b, unsigned c) {
#if __has_builtin(__builtin_amdgcn_sad_u8)
    return __builtin_amdgcn_sad_u8(a, b, c);
#else
    unsigned r;
    asm("v_sad_u8 %0, %1, %2, %3" : "=v"(r) : "v"(a), "v"(b), "v"(c));
    return r;
#endif
}

// ---------------------------------------------------------------------------
// Kernel 1: weight scale (global max|w| -> power-of-2 snap) + quantize+bias,
// relayout OIHW -> [cout][ky*3+kx][cin] u8. Single block.
// ---------------------------------------------------------------------------
__global__ __launch_bounds__(256) void qw_kernel(const float* __restrict__ w,
                                                 unsigned char* __restrict__ qw) {
    __shared__ float red[256];
    float m = 0.0f;
    for (int i = threadIdx.x; i < COUT * CIN * 9; i += 256) m = fmaxf(m, fabsf(w[i]));
    red[threadIdx.x] = m;
    __syncthreads();
    for (int s = 128; s > 0; s >>= 1) {
        if ((int)threadIdx.x < s) red[threadIdx.x] = fmaxf(red[threadIdx.x], red[threadIdx.x + s]);
        __syncthreads();
    }
    float scale = fmaxf(red[0] * (1.0f / 127.0f), 1e-8f);
    scale = exp2f(rintf(log2f(scale)));     // snap to power of 2
    float inv = 1.0f / scale;               // exact (power of 2)
    for (int i = threadIdx.x; i < COUT * CIN * 9; i += 256) {
        // input index: ((o*32 + ci)*3 + ky)*3 + kx
        int kx = i % 3;
        int t  = i / 3;
        int ky = t % 3;  t /= 3;
        int ci = t % CIN;
        int o  = t / CIN;
        float q = rintf(fminf(fmaxf(w[i] * inv, -127.0f), 127.0f));
        qw[(o * 9 + ky * 3 + kx) * CIN + ci] = (unsigned char)((int)q + 127); // bias -> [0,254]
    }
}

// ---------------------------------------------------------------------------
// Kernel 2: quantize x (scale 2^-7) + bias, transpose NCHW f32 -> NHWC u8.
// One block per (n,h) row: 32 c-planes x 128 w, via LDS transpose.
// ---------------------------------------------------------------------------
__global__ __launch_bounds__(256) void qx_kernel(const float* __restrict__ x,
                                                 unsigned char* __restrict__ qx) {
    __shared__ __align__(16) unsigned char tile[WW * CIN];   // 4096 B
    int nh = blockIdx.x;                 // n*128 + h
    int n  = nh >> 7;
    int h  = nh & 127;
    const float* src = x + (size_t)n * CIN * HH * WW + (size_t)h * WW;  // + c*16384 + w
    for (int k = 0; k < 16; ++k) {
        int e  = k * 256 + threadIdx.x;  // [0,4096)
        int c  = e >> 7;
        int wv = e & 127;                // consecutive tid -> consecutive w (coalesced)
        float v = src[(size_t)c * (HH * WW) + wv] * 128.0f;
        v = rintf(fminf(fmaxf(v, -127.0f), 127.0f));
        tile[wv * CIN + c] = (unsigned char)((int)v + 127);
    }
    __syncthreads();
    // 4096 B = 256 x b128, one per thread, coalesced
    uint4* dst = (uint4*)(qx + (size_t)nh * (WW * CIN));
    dst[threadIdx.x] = ((const uint4*)tile)[threadIdx.x];
}

// ---------------------------------------------------------------------------
// Kernel 3: SAD conv. Block = 16x16 output pixels (all 32 couts), grid (64,16).
// x tile (18x18x32 u8, halo = biased zero 0x7F) staged into LDS with the
// CDNA5 async-to-LDS path; weights read wave-uniformly (SMEM/constant cache).
// Per thread: 32 accumulators, 9 kpos x 32 cout x 8 v_sad_u8.
// ---------------------------------------------------------------------------
__global__ __launch_bounds__(256) void sad_kernel(const unsigned char* __restrict__ qx,
                                                  const unsigned char* __restrict__ qw,
                                                  float* __restrict__ out) {
    __shared__ __align__(16) unsigned char xt[18 * 18 * 32];   // 10368 B, row pitch 576
    const int n   = blockIdx.y;
    const int ty  = blockIdx.x >> 3;
    const int tx  = blockIdx.x & 7;
    const int tid = threadIdx.x;

    // Prefill halo/pad with biased-zero (127) so out-of-image taps contribute |0 - qw|.
    {
        uint4 fill = make_uint4(0x7F7F7F7Fu, 0x7F7F7F7Fu, 0x7F7F7F7Fu, 0x7F7F7F7Fu);
        uint4* p = (uint4*)xt;
        for (int i = tid; i < (18 * 18 * 32) / 16; i += 256) p[i] = fill;
    }
    __syncthreads();

    // Async-copy valid 16B chunks global->LDS: 18 rows x 36 chunks (each chunk
    // lies entirely within one input column's 32-byte channel vector).
    for (int q = tid; q < 18 * 36; q += 256) {
        int r   = q / 36;              // tile row 0..17
        int cb  = (q % 36) * 16;       // byte offset within tile row
        int col = cb >> 5;             // tile col 0..17
        int iy  = ty * 16 - 1 + r;
        int ix  = tx * 16 - 1 + col;
        if ((unsigned)iy < (unsigned)HH && (unsigned)ix < (unsigned)WW) {
            unsigned lds = (unsigned)(size_t)(xt + r * 576 + cb);  // low 32 bits = LDS offset
            unsigned long long g = (unsigned long long)(size_t)
                (qx + (((size_t)(n * HH + iy) * WW + ix) * CIN) + (cb & 31));
            asm volatile("global_load_async_to_lds_b128 %0, %1, off"
                         :: "v"(lds), "v"(g) : "memory");
        }
    }
    asm volatile("s_wait_asynccnt 0" ::: "memory");
    __syncthreads();

    const int oy = tid >> 4;
    const int ox = tid & 15;

    unsigned acc[COUT];
#pragma unroll
    for (int i = 0; i < COUT; ++i) acc[i] = 0u;

    const unsigned* __restrict__ w32 = (const unsigned*)qw;   // wave-uniform accesses

#pragma unroll 1
    for (int kp = 0; kp < 9; ++kp) {
        int ky = kp / 3, kx = kp - 3 * ky;
        int xo = (((oy + ky) * 18) + (ox + kx)) * 32;
        uint4 a0 = *(const uint4*)(xt + xo);
        uint4 a1 = *(const uint4*)(xt + xo + 16);
#pragma unroll
        for (int co = 0; co < COUT; ++co) {
            const unsigned* wp = w32 + (co * 9 + kp) * 8;     // uniform -> s_load
            unsigned s = acc[co];
            s = sad8(a0.x, wp[0], s);
            s = sad8(a0.y, wp[1], s);
            s = sad8(a0.z, wp[2], s);
            s = sad8(a0.w, wp[3], s);
            s = sad8(a1.x, wp[4], s);
            s = sad8(a1.y, wp[5], s);
            s = sad8(a1.z, wp[6], s);
            s = sad8(a1.w, wp[7], s);
            acc[co] = s;
        }
    }

    // out[n][co][oyg][oxg] = -SAD * 2^-7
    int oyg = ty * 16 + oy;
    int oxg = tx * 16 + ox;
    size_t base = (size_t)n * COUT * HH * WW + (size_t)oyg * WW + oxg;
#pragma unroll
    for (int co = 0; co < COUT; ++co)
        out[base + (size_t)co * (HH * WW)] = -(float)(int)acc[co] * 0.0078125f;
}

// ---------------------------------------------------------------------------
extern "C" void kernel_launch(void* const* d_in, const int* in_sizes, int n_in,
                              void* d_out, int out_size, void* d_ws, size_t ws_size,
                              hipStream_t stream) {
    const float* x = (const float*)d_in[0];       // [16,32,128,128]
    const float* w = (const float*)d_in[1];       // [32,32,3,3]
    float* out = (float*)d_out;                   // [16,32,128,128]

    unsigned char* qx = (unsigned char*)d_ws;                 // QX_BYTES
    unsigned char* qw = (unsigned char*)d_ws + QX_BYTES;      // QW_BYTES

    qw_kernel<<<1, 256, 0, stream>>>(w, qw);
    qx_kernel<<<NB * HH, 256, 0, stream>>>(x, qx);
    sad_kernel<<<dim3((HH / 16) * (WW / 16), NB), 256, 0, stream>>>(qx, qw, out);
    (void)in_sizes; (void)n_in; (void)out_size; (void)ws_size;
}